// FuzzySignedConjunction_13039520711341
// MI455X (gfx1250) — compile-verified
//
#include <hip/hip_runtime.h>
#include <math.h>

#define B_DIM 2048
#define I_DIM 512
#define O_DIM 512
#define DEG   7          // powers x^1..x^7; P_0 folded into bias
#define NNODES 8         // Chebyshev interpolation nodes -> degree-7 poly

typedef __attribute__((ext_vector_type(2))) float v2f;
typedef __attribute__((ext_vector_type(4))) float v4f;
typedef __attribute__((ext_vector_type(8))) float v8f;
typedef int i32x4 __attribute__((vector_size(16)));   // matches builtin param

// Async global->LDS copy path (CDNA5 GLOBAL_LOAD_ASYNC_TO_LDS_B128, ASYNCcnt)
#if defined(__gfx1250__) && __has_builtin(__builtin_amdgcn_global_load_async_to_lds_b128)
#define HAVE_ASYNC_LDS 1
#else
#define HAVE_ASYNC_LDS 0
#endif

#if HAVE_ASYNC_LDS
#define ASYNC_CP16(dst_lds, src_glb)                                          \
    __builtin_amdgcn_global_load_async_to_lds_b128(                           \
        (__attribute__((address_space(1))) i32x4*)(src_glb),                  \
        (__attribute__((address_space(3))) i32x4*)(dst_lds), 0, 0)
static __device__ __forceinline__ void wait_async0() {
#if __has_builtin(__builtin_amdgcn_s_wait_asynccnt)
    __builtin_amdgcn_s_wait_asynccnt(0);
#else
    asm volatile("s_wait_asynccnt 0x0" ::: "memory");
#endif
}
#endif

// Workspace layout:
//   P0   : [I][O]            plain        (only feeds bias reduction)
//   Pint : [7][I/2][O][2]    row-pair interleaved: {P_k[i][o], P_k[i+1][o]}
//   bias : [O]
// The interleave makes each WMMA B-fragment one aligned ds_load_b64.

// ---------------------------------------------------------------------------
// Kernel 1: per-(i,o) degree-7 polynomial fit of g(x) = log((1-a) - c*x)
// at Chebyshev nodes on [0,1]; Lagrange -> monomial basis, in f64.
// ---------------------------------------------------------------------------
__global__ void fuzzy_coeff_kernel(const float* __restrict__ w_logits,
                                   const float* __restrict__ s_logits,
                                   float* __restrict__ P0,
                                   float* __restrict__ Pint)
{
    int idx = blockIdx.x * blockDim.x + threadIdx.x;
    if (idx >= I_DIM * O_DIM) return;
    const int i = idx / O_DIM;
    const int o = idx % O_DIM;

    const double PI = 3.14159265358979323846;
    double w = 1.0 / (1.0 + exp(-(double)w_logits[idx]));
    double s = 1.0 / (1.0 + exp(-(double)s_logits[idx]));
    double a = w * s;              // factor = (1 - a) - c*x
    double c = w * (1.0 - 2.0 * s);

    double xs[NNODES], gs[NNODES];
    #pragma unroll
    for (int j = 0; j < NNODES; ++j) {
        double t = cos((2.0 * j + 1.0) * PI / (2.0 * NNODES));
        double x = 0.5 + 0.5 * t;                       // Chebyshev node in [0,1]
        double f = 1.0 - a - c * x;                     // factor > 0 always
        if (f < 1e-30) f = 1e-30;
        xs[j] = x;
        gs[j] = log(f);
    }

    double coef[NNODES];
    #pragma unroll
    for (int k = 0; k < NNODES; ++k) coef[k] = 0.0;
    #pragma unroll
    for (int j = 0; j < NNODES; ++j) {
        double poly[NNODES];
        poly[0] = 1.0;
        #pragma unroll
        for (int t = 1; t < NNODES; ++t) poly[t] = 0.0;
        int deg = 0;
        double denom = 1.0;
        #pragma unroll
        for (int m = 0; m < NNODES; ++m) {
            if (m == j) continue;
            ++deg;
            for (int t = deg; t >= 1; --t) poly[t] = poly[t - 1] - xs[m] * poly[t];
            poly[0] = -xs[m] * poly[0];
            denom *= (xs[j] - xs[m]);
        }
        double sc = gs[j] / denom;
        #pragma unroll
        for (int t = 0; t < NNODES; ++t) coef[t] += sc * poly[t];
    }

    P0[idx] = (float)coef[0];
    const size_t pairBase = ((size_t)(i >> 1) * O_DIM + o) * 2 + (i & 1);
    #pragma unroll
    for (int k = 1; k < NNODES; ++k)
        Pint[(size_t)(k - 1) * I_DIM * O_DIM + pairBase] = (float)coef[k];
}

// ---------------------------------------------------------------------------
// Kernel 2: bias[o] = sum_i P0[i][o]
// ---------------------------------------------------------------------------
__global__ void fuzzy_bias_kernel(const float* __restrict__ P0,
                                  float* __restrict__ bias)
{
    int o = blockIdx.x * blockDim.x + threadIdx.x;
    if (o >= O_DIM) return;
    float sum = 0.f;
    for (int i = 0; i < I_DIM; ++i) sum += P0[(size_t)i * O_DIM + o];
    bias[o] = sum;
}

// ---------------------------------------------------------------------------
// Kernel 3: out[b,o] = exp( bias[o] + sum_{k=1..7} sum_i x[b,i]^k * P_k[i,o] )
// fp32 GEMM on the matrix pipe (V_WMMA_F32_16X16X4_F32), double-buffered LDS,
// async global->LDS prefetch of block i+1 under block i's 224 WMMAs,
// B-fragments as single aligned ds_load_b64 via row-pair interleaving.
// ---------------------------------------------------------------------------
#define TILE_B   128
#define TILE_O   64
#define KBLK     16
#define XSTRIDE  20   // 16 + 4 pad floats: 80B rows, 16B aligned, spread banks
#define PSTRIDE2 160  // 128 data + 32 pad floats: 640B rows, 16B aligned;
                      // stride mod 64 == 32 -> lane16-31 fragment row lands on
                      // banks +32 vs lanes 0-15 -> conflict-free b64 reads

__global__ __launch_bounds__(128)
void fuzzy_gemm_kernel(const float* __restrict__ x,
                       const float* __restrict__ Pint,   // [7][I/2][O][2]
                       const float* __restrict__ bias,
                       float* __restrict__ out)
{
    __shared__ float lds_x[2][TILE_B * XSTRIDE];
    __shared__ float lds_p[2][DEG * (KBLK / 2) * PSTRIDE2];

    const int tid  = threadIdx.x;
    const int lane = tid & 31;
    const int wv   = tid >> 5;        // wave 0..3
    const int half = lane >> 4;       // 0: K{0,1}, 1: K{2,3} per A-layout
    const int l16  = lane & 15;

    const int b0 = blockIdx.x * TILE_B;
    const int o0 = blockIdx.y * TILE_O;

    v8f acc[2][4] = {};               // [substrip][o-subtile]

    // stage one KBLK-deep block of x and P_1..P_7 into lds buffer `buf`
    auto stage = [&](int i0, int buf) {
        const float* xsrc = x + (size_t)(b0 + tid) * I_DIM + i0;
        float*       xdst = &lds_x[buf][tid * XSTRIDE];
        #pragma unroll
        for (int ch = 0; ch < 4; ++ch) {
#if HAVE_ASYNC_LDS
            ASYNC_CP16(xdst + ch * 4, xsrc + ch * 4);
#else
            *(v4f*)(xdst + ch * 4) = *(const v4f*)(xsrc + ch * 4);
#endif
        }
        // 7k x 8 row-pairs x 8 b128 chunks = 448 jobs over 128 threads
        #pragma unroll
        for (int it = 0; it < 4; ++it) {
            int job = it * 128 + tid;
            if (job < DEG * 8 * 8) {
                int k  = job >> 6;            // 0..6 -> P_{k+1}
                int q  = (job >> 3) & 7;      // row pair within block
                int ch = job & 7;             // 16B chunk within 64-o slice
                const float* src = Pint + (size_t)k * I_DIM * O_DIM
                                 + ((size_t)(i0 / 2 + q) * O_DIM + o0) * 2
                                 + ch * 4;
                float* dst = &lds_p[buf][(k * 8 + q) * PSTRIDE2 + ch * 4];
#if HAVE_ASYNC_LDS
                ASYNC_CP16(dst, src);
#else
                *(v4f*)dst = *(const v4f*)src;
#endif
            }
        }
    };

    stage(0, 0);
    int cur = 0;
    for (int i0 = 0; i0 < I_DIM; i0 += KBLK) {
#if HAVE_ASYNC_LDS
        wait_async0();                 // my wave's copies into buf[cur] done
#endif
        __syncthreads();               // everyone's copies done; buf[cur^1] free
        if (i0 + KBLK < I_DIM)
            stage(i0 + KBLK, cur ^ 1); // prefetch next block under compute

        // ---- compute on buf[cur]: 4 K-steps, 7 powers, 4 o-subtiles, 2 strips
        const float* lx = lds_x[cur];
        const float* lp = lds_p[cur];
        #pragma unroll
        for (int t = 0; t < 4; ++t) {
            const int c0 = t * 4 + half * 2;   // A: VGPR0=K{0|2}, VGPR1=K{1|3}
            v2f a0 = *(const v2f*)(lx + (wv * 32 +  0 + l16) * XSTRIDE + c0);
            v2f a1 = *(const v2f*)(lx + (wv * 32 + 16 + l16) * XSTRIDE + c0);
            v2f p0 = a0, p1 = a1;
            #pragma unroll
            for (int k = 0; k < DEG; ++k) {
                if (k) { p0 *= a0; p1 *= a1; }     // p = a^(k+1)
                // fragment rows {4t+2*half, 4t+2*half+1} sit interleaved:
                // one aligned b64 per o-subtile, no repacking movs
                const float* prow =
                    lp + (k * 8 + t * 2 + half) * PSTRIDE2 + l16 * 2;
                #pragma unroll
                for (int n = 0; n < 4; ++n) {
                    v2f bfrag = *(const v2f*)(prow + n * 32);
                    acc[0][n] = __builtin_amdgcn_wmma_f32_16x16x4_f32(
                        false, p0, false, bfrag, (short)0, acc[0][n], false, false);
                    acc[1][n] = __builtin_amdgcn_wmma_f32_16x16x4_f32(
                        false, p1, false, bfrag, (short)0, acc[1][n], false, false);
                }
            }
        }
        cur ^= 1;
    }

    // ---- epilogue: out = exp(acc + bias). C/D layout: VGPR r, lane l ->
    // row = r + (l>=16 ? 8 : 0), col = l&15 ----
    #pragma unroll
    for (int s = 0; s < 2; ++s) {
        #pragma unroll
        for (int n = 0; n < 4; ++n) {
            const int col = o0 + n * 16 + l16;
            const float bb = bias[col];
            #pragma unroll
            for (int r = 0; r < 8; ++r) {
                const int row = b0 + wv * 32 + s * 16 + half * 8 + r;
                out[(size_t)row * O_DIM + col] = __expf(acc[s][n][r] + bb);
            }
        }
    }
}

// ---------------------------------------------------------------------------
extern "C" void kernel_launch(void* const* d_in, const int* in_sizes, int n_in,
                              void* d_out, int out_size, void* d_ws, size_t ws_size,
                              hipStream_t stream)
{
    const float* x  = (const float*)d_in[0];   // (B, I)
    const float* wl = (const float*)d_in[1];   // (I, O)
    const float* sl = (const float*)d_in[2];   // (I, O)
    float* P0   = (float*)d_ws;                           // [I][O]
    float* Pint = P0 + (size_t)I_DIM * O_DIM;             // [7][I/2][O][2]
    float* bias = P0 + (size_t)NNODES * I_DIM * O_DIM;    // [O]
    float* out  = (float*)d_out;                          // (B, O)

    fuzzy_coeff_kernel<<<(I_DIM * O_DIM + 255) / 256, 256, 0, stream>>>(wl, sl, P0, Pint);
    fuzzy_bias_kernel<<<(O_DIM + 255) / 256, 256, 0, stream>>>(P0, bias);

    dim3 grid(B_DIM / TILE_B, O_DIM / TILE_O);   // 16 x 8 = 128 workgroups
    fuzzy_gemm_kernel<<<grid, 128, 0, stream>>>(x, Pint, bias, out);
}